// FieldOptimizerShifts_40157944217684
// MI455X (gfx1250) — compile-verified
//
#include <hip/hip_runtime.h>
#include <math.h>

// ---------------- problem constants (match reference) ----------------
#define BDIM 2
#define CDIM 32
#define HDIM 256
#define WDIM 256
#define NIMG (BDIM*CDIM)              // 64 images
#define NTOT (NIMG*HDIM*WDIM)         // 4194304 elements per field
#define NUM_STEPS 50

#define ALPHA  0.1f
#define BETA_  0.1f
#define VV     0.1f
#define LR     0.05f
#define GAMMA_ 0.07f
#define AB1    0.9f
#define AB2    0.999f
#define AEPS   1e-8f

#define RED_BLOCKS  512
#define RED_THREADS 256

typedef __attribute__((ext_vector_type(2))) float v2f;
typedef __attribute__((ext_vector_type(8))) float v8f;

// Async memory->LDS copy (CDNA5 GLOBAL_LOAD_ASYNC_TO_LDS_B32, GV mode).
// ldsaddr = LDS byte address (low 32 bits of flat shared address, ISA §10.2),
// gaddr   = 64-bit global byte address. Tracked by ASYNCcnt.
__device__ __forceinline__ void async_load_b32(unsigned ldsaddr,
                                               unsigned long long gaddr) {
  asm volatile("global_load_async_to_lds_b32 %0, %1, off"
               :: "v"(ldsaddr), "v"(gaddr) : "memory");
}
__device__ __forceinline__ void wait_asynccnt0() {
  asm volatile("s_wait_asynccnt 0x0" ::: "memory");
}
__device__ __forceinline__ unsigned lds_addr_of(const void* p) {
  return (unsigned)(unsigned long long)(uintptr_t)p;  // addr[31:0] == LDS offset
}

// ---------------- init: copy phi0 into ping buffer, zero Adam state ----------------
__global__ void k_init(const float* __restrict__ p0x, const float* __restrict__ p0y,
                       float* __restrict__ px, float* __restrict__ py,
                       float* __restrict__ mx, float* __restrict__ my,
                       float* __restrict__ vx, float* __restrict__ vy) {
  int i = blockIdx.x * blockDim.x + threadIdx.x;
  if (i < NTOT) {
    px[i] = p0x[i]; py[i] = p0y[i];
    mx[i] = 0.f; my[i] = 0.f; vx[i] = 0.f; vy[i] = 0.f;
  }
}

// ---------------- reduction stage 1: sum(px^2), sum(py^2) via WMMA ----------------
// B = all-ones => D = A*1 + C accumulates row-sums: one v_wmma folds 64 squared
// values (2 VGPRs x 32 lanes) into the f32 accumulator. Counted uniform loop
// (exact division) keeps EXEC all-ones around the WMMAs with a scalar branch.
__global__ __launch_bounds__(RED_THREADS) void k_reduce1(
    const float* __restrict__ px, const float* __restrict__ py,
    float* __restrict__ partials) {
  const int tid = threadIdx.x;
  const int gid = blockIdx.x * blockDim.x + tid;
  const int nthreads = gridDim.x * blockDim.x;       // 131072
  const int n2 = NTOT / 2;                           // 2097152
  const int iters = n2 / nthreads;                   // 16, uniform
  const float2* px2 = (const float2*)px;
  const float2* py2 = (const float2*)py;

  v2f ones = {1.0f, 1.0f};
  v8f accx = {};
  v8f accy = {};
  int i = gid;
  for (int k = 0; k < iters; ++k) {
    float2 a = px2[i];
    float2 b = py2[i];
    v2f av = {a.x * a.x, a.y * a.y};
    v2f bv = {b.x * b.x, b.y * b.y};
    accx = __builtin_amdgcn_wmma_f32_16x16x4_f32(false, av, false, ones,
                                                 (short)0, accx, false, false);
    accy = __builtin_amdgcn_wmma_f32_16x16x4_f32(false, bv, false, ones,
                                                 (short)0, accy, false, false);
    i += nthreads;
  }
  // D[i][j] = rowSum_i (replicated 16x across columns). Per-lane sum of 8 regs
  // then cross-lane sum over wave32 yields 16 * waveSum.
  float sx = 0.f, sy = 0.f;
#pragma unroll
  for (int k = 0; k < 8; ++k) { sx += accx[k]; sy += accy[k]; }
  for (int off = 16; off > 0; off >>= 1) {
    sx += __shfl_xor(sx, off);
    sy += __shfl_xor(sy, off);
  }
  __shared__ float lsx[8], lsy[8];
  const int wave = tid >> 5, lane = tid & 31;
  if (lane == 0) { lsx[wave] = sx; lsy[wave] = sy; }
  __syncthreads();
  if (tid == 0) {
    float tx = 0.f, ty = 0.f;
    for (int w = 0; w < (RED_THREADS >> 5); ++w) { tx += lsx[w]; ty += lsy[w]; }
    partials[2 * blockIdx.x]     = tx * (1.0f / 16.0f);
    partials[2 * blockIdx.x + 1] = ty * (1.0f / 16.0f);
  }
}

// ---------------- reduction stage 2: scalars for SSB gradient ----------------
__global__ __launch_bounds__(RED_THREADS) void k_reduce2(
    const float* __restrict__ partials, float* __restrict__ scal) {
  __shared__ float sxs[RED_THREADS], sys[RED_THREADS];
  const int tid = threadIdx.x;
  float sx = 0.f, sy = 0.f;
  for (int i = tid; i < RED_BLOCKS; i += RED_THREADS) {
    sx += partials[2 * i];
    sy += partials[2 * i + 1];
  }
  sxs[tid] = sx; sys[tid] = sy;
  __syncthreads();
  for (int s = RED_THREADS / 2; s > 0; s >>= 1) {
    if (tid < s) { sxs[tid] += sxs[tid + s]; sys[tid] += sys[tid + s]; }
    __syncthreads();
  }
  if (tid == 0) {
    float Sx = sxs[0], Sy = sys[0];
    float rx = sqrtf(Sx), ry = sqrtf(Sy);
    float pn = rx + ry;
    float d  = pn * pn - VV * VV;
    float sg = (d > 0.f) ? 1.f : ((d < 0.f) ? -1.f : 0.f);
    // d(beta*|pn^2 - V^2|)/dphi = 2*beta*sign*pn/sqrt(S_field) * phi
    scal[0] = (rx > 0.f) ? (2.f * BETA_ * sg * pn / rx) : 0.f;
    scal[1] = (ry > 0.f) ? (2.f * BETA_ * sg * pn / ry) : 0.f;
  }
}

// ---------------- stencil helpers: cross-correlation with Sobel/8 ----------------
__device__ __forceinline__ float corr_dx(const float* f, int st, int r, int c) {
  return ((f[(r - 1) * st + c + 1] - f[(r - 1) * st + c - 1])
        + 2.f * (f[r * st + c + 1] - f[r * st + c - 1])
        + (f[(r + 1) * st + c + 1] - f[(r + 1) * st + c - 1])) * 0.125f;
}
__device__ __forceinline__ float corr_dy(const float* f, int st, int r, int c) {
  return ((f[(r + 1) * st + c - 1] - f[(r - 1) * st + c - 1])
        + 2.f * (f[(r + 1) * st + c] - f[(r - 1) * st + c])
        + (f[(r + 1) * st + c + 1] - f[(r - 1) * st + c + 1])) * 0.125f;
}

// ---------------- fused gradient + Adam step (one pass per step) ----------------
// Halo tiles are staged with CDNA5 async memory->LDS copies (ASYNCcnt); the
// out-of-image halo entries are zero-filled with plain DS stores (disjoint LDS
// locations). Kinetic conv-of-conv is fused via halo-2 phi + halo-1 g tiles.
__global__ __launch_bounds__(256) void k_step(
    const float* __restrict__ S,
    const float* __restrict__ pxin, const float* __restrict__ pyin,
    float* __restrict__ pxout, float* __restrict__ pyout,
    float* __restrict__ mx, float* __restrict__ my,
    float* __restrict__ vx, float* __restrict__ vy,
    const float* __restrict__ scal,
    float bc1, float bc2, int last, float* __restrict__ outSp) {
  __shared__ float sPx[36 * 37];   // phi_x tile with halo 2
  __shared__ float sPy[36 * 37];   // phi_y tile with halo 2
  __shared__ float sS [33 * 34];   // S tile with +1 halo (right/bottom)
  __shared__ float g1x[34 * 35];   // corr(phi_x, kdx) on halo-1 region
  __shared__ float g2x[34 * 35];   // corr(phi_x, kdy)
  __shared__ float g1y[34 * 35];   // corr(phi_y, kdx)
  __shared__ float g2y[34 * 35];   // corr(phi_y, kdy)

  const int tx0 = blockIdx.x * 32;
  const int ty0 = blockIdx.y * 32;
  const int img = blockIdx.z;
  const long imgOff = (long)img * HDIM * WDIM;
  const int tid = threadIdx.x;

  const float* basePx = pxin + imgOff;
  const float* basePy = pyin + imgOff;
  const float* baseS  = S + imgOff;

  // phi halo-2 tiles: async memory->LDS for in-image entries, DS-store zeros
  // for the zero-padding of the inner conv.
  for (int idx = tid; idx < 36 * 36; idx += 256) {
    int r = idx / 36, c = idx % 36;
    int gy = ty0 - 2 + r, gx = tx0 - 2 + c;
    if (gy >= 0 && gy < HDIM && gx >= 0 && gx < WDIM) {
      unsigned long long off = (unsigned long long)(gy * WDIM + gx) * 4ull;
      async_load_b32(lds_addr_of(&sPx[r * 37 + c]),
                     (unsigned long long)(uintptr_t)basePx + off);
      async_load_b32(lds_addr_of(&sPy[r * 37 + c]),
                     (unsigned long long)(uintptr_t)basePy + off);
    } else {
      sPx[r * 37 + c] = 0.f;
      sPy[r * 37 + c] = 0.f;
    }
  }
  // S tile with +1 halo (forward differences need S[y+1], S[x+1])
  for (int idx = tid; idx < 33 * 33; idx += 256) {
    int r = idx / 33, c = idx % 33;
    int gy = ty0 + r, gx = tx0 + c;
    if (gy < HDIM && gx < WDIM) {
      unsigned long long off = (unsigned long long)(gy * WDIM + gx) * 4ull;
      async_load_b32(lds_addr_of(&sS[r * 34 + c]),
                     (unsigned long long)(uintptr_t)baseS + off);
    } else {
      sS[r * 34 + c] = 0.f;
    }
  }
  wait_asynccnt0();   // ASYNCcnt: async tile copies complete
  __syncthreads();    // (compiler adds DScnt wait for the zero stores)

  // inner Sobel correlations on tile + 1 halo. Positions outside the image are
  // set to 0: the *outer* conv zero-pads its input (two-stage conv semantics).
  for (int idx = tid; idx < 34 * 34; idx += 256) {
    int rr = idx / 34, cc = idx % 34;
    int gy = ty0 - 1 + rr, gx = tx0 - 1 + cc;
    float a = 0.f, b = 0.f, c2 = 0.f, d = 0.f;
    if (gy >= 0 && gy < HDIM && gx >= 0 && gx < WDIM) {
      int r = rr + 1, c = cc + 1;    // index into 36x36 phi tiles
      a  = corr_dx(sPx, 37, r, c);
      b  = corr_dy(sPx, 37, r, c);
      c2 = corr_dx(sPy, 37, r, c);
      d  = corr_dy(sPy, 37, r, c);
    }
    g1x[rr * 35 + cc] = a;
    g2x[rr * 35 + cc] = b;
    g1y[rr * 35 + cc] = c2;
    g2y[rr * 35 + cc] = d;
  }
  __syncthreads();

  const float cx    = scal[0];
  const float cy    = scal[1];
  const float twoN  = 2.0f / (float)NTOT;                 // coupling grad scale
  const float kinC  = 2.0f * ALPHA / (float)NTOT;         // kinetic grad scale
  const float voidC = 2.0f * GAMMA_ * GAMMA_;             // gamma^2 void grad

  for (int idx = tid; idx < 32 * 32; idx += 256) {
    int r = idx / 32, c = idx % 32;
    int gy = ty0 + r, gx = tx0 + c;
    long p = imgOff + (long)gy * WDIM + gx;

    float pxv = sPx[(r + 2) * 37 + (c + 2)];
    float pyv = sPy[(r + 2) * 37 + (c + 2)];
    float Sv  = sS[r * 34 + c];
    float dxv = (gx == WDIM - 1) ? 0.f : (sS[r * 34 + c + 1] - Sv);
    float dyv = (gy == HDIM - 1) ? 0.f : (sS[(r + 1) * 34 + c] - Sv);
    float maskv = (Sv < 0.05f) ? 1.f : 0.f;
    float rc = pxv * dxv + pyv * dyv;       // Sp - S

    // kinetic grad: transpose of zero-padded correlation with antisymmetric
    // Sobel == minus correlation => grad = -(2a/N)(corr(g_dx,kdx)+corr(g_dy,kdy))
    float gradx = twoN * rc * dxv
                - kinC * (corr_dx(g1x, 35, r + 1, c + 1) + corr_dy(g2x, 35, r + 1, c + 1))
                + cx * pxv + voidC * maskv * pxv;
    float grady = twoN * rc * dyv
                - kinC * (corr_dx(g1y, 35, r + 1, c + 1) + corr_dy(g2y, 35, r + 1, c + 1))
                + cy * pyv + voidC * maskv * pyv;

    float mxv = AB1 * mx[p] + (1.f - AB1) * gradx;
    float myv = AB1 * my[p] + (1.f - AB1) * grady;
    float vxv = AB2 * vx[p] + (1.f - AB2) * gradx * gradx;
    float vyv = AB2 * vy[p] + (1.f - AB2) * grady * grady;
    mx[p] = mxv; my[p] = myv; vx[p] = vxv; vy[p] = vyv;

    pxout[p] = pxv - LR * (mxv / bc1) / (sqrtf(vxv / bc2) + AEPS);
    pyout[p] = pyv - LR * (myv / bc1) / (sqrtf(vyv / bc2) + AEPS);
    if (last) outSp[p] = Sv + rc;   // S' from pre-update phi of the last step
  }
}

// ---------------- host orchestration ----------------
extern "C" void kernel_launch(void* const* d_in, const int* in_sizes, int n_in,
                              void* d_out, int out_size, void* d_ws, size_t ws_size,
                              hipStream_t stream) {
  const float* S   = (const float*)d_in[0];
  const float* p0x = (const float*)d_in[1];
  const float* p0y = (const float*)d_in[2];

  float* out  = (float*)d_out;
  float* oSp  = out;                       // output slot 0: S'
  float* oPxA = out + 1 * (size_t)NTOT;    // slot 1: phi_x post  (ping A)
  float* oPyA = out + 2 * (size_t)NTOT;    // slot 2: phi_y post  (ping A)
  float* oPxB = out + 3 * (size_t)NTOT;    // slot 3: phi_x pre   (ping B)
  float* oPyB = out + 4 * (size_t)NTOT;    // slot 4: phi_y pre   (ping B)

  float* wsf      = (float*)d_ws;
  float* mx       = wsf;
  float* my       = wsf + 1 * (size_t)NTOT;
  float* vx       = wsf + 2 * (size_t)NTOT;
  float* vy       = wsf + 3 * (size_t)NTOT;
  float* partials = wsf + 4 * (size_t)NTOT;
  float* scal     = partials + 2 * RED_BLOCKS;

  k_init<<<NTOT / 256, 256, 0, stream>>>(p0x, p0y, oPxA, oPyA, mx, my, vx, vy);

  // Ping-pong between out-slots (1,2) and (3,4). 50 steps is even, so the
  // final phi lands in slots (1,2) and the step-50 *input* phi remains in
  // slots (3,4) == exactly the required (post, pre) output layout. Zero copies.
  float* pix = oPxA; float* piy = oPyA;
  float* pox = oPxB; float* poy = oPyB;
  double b1t = 1.0, b2t = 1.0;
  dim3 grid(WDIM / 32, HDIM / 32, NIMG);
  for (int t = 1; t <= NUM_STEPS; ++t) {
    b1t *= (double)AB1;
    b2t *= (double)AB2;
    float bc1 = (float)(1.0 - b1t);
    float bc2 = (float)(1.0 - b2t);
    k_reduce1<<<RED_BLOCKS, RED_THREADS, 0, stream>>>(pix, piy, partials);
    k_reduce2<<<1, RED_THREADS, 0, stream>>>(partials, scal);
    int last = (t == NUM_STEPS) ? 1 : 0;
    k_step<<<grid, 256, 0, stream>>>(S, pix, piy, pox, poy,
                                     mx, my, vx, vy, scal, bc1, bc2, last, oSp);
    float* tx = pix; pix = pox; pox = tx;
    float* ty = piy; piy = poy; poy = ty;
  }
}